// StrengthNet_40699110097065
// MI455X (gfx1250) — compile-verified
//
#include <hip/hip_runtime.h>
#include <hip/hip_bf16.h>

// ---------------------------------------------------------------------------
// StrengthNet forward for MI455X (gfx1250, wave32).
// Memory-bound (~90MB traffic, ~1.4 GFLOP) -> FP32 throughout.
// Layer-1 GEMM on the matrix pipe via V_WMMA_F32_16X16X4_F32, computed as
// H^T = W1^T * X^T so tokens land on lanes (cheap head reduction: one
// xor-16 shuffle instead of a 4-level tree per C register).
// ---------------------------------------------------------------------------

#define FEAT 6
#define DHID 32
#define SCALE_F 173.71779276130070f   // 400 / ln(10)

typedef __attribute__((ext_vector_type(2))) float v2f;
typedef __attribute__((ext_vector_type(8))) float v8f;

// monotone float -> u32 ordering key (0 sorts below every finite key => "-inf")
__device__ __forceinline__ unsigned int f32_key(float f) {
    unsigned int u = __float_as_uint(f);
    return (u & 0x80000000u) ? ~u : (u | 0x80000000u);
}
__device__ __forceinline__ float f32_unkey(unsigned int k) {
    unsigned int u = (k & 0x80000000u) ? (k ^ 0x80000000u) : ~k;
    return __uint_as_float(u);
}

__global__ void k_init(unsigned int* __restrict__ zmaxk,
                       float* __restrict__ num, float* __restrict__ den, int G) {
    int g = blockIdx.x * blockDim.x + threadIdx.x;
    if (g < G) { zmaxk[g] = 0u; num[g] = 0.0f; den[g] = 0.0f; }
}

// One wave = 4 tiles of 16 tokens = 64 tokens.  4 WMMAs per tile.
#define TILES_PER_WAVE 4
#define TOK_PER_WAVE   (16 * TILES_PER_WAVE)

__global__ void k_mlp(const float* __restrict__ x,
                      const float* __restrict__ W1, const float* __restrict__ b1,
                      const float* __restrict__ wr, const float* __restrict__ brp,
                      const float* __restrict__ wz, const float* __restrict__ bzp,
                      const int*   __restrict__ seg,
                      float* __restrict__ rbuf, float* __restrict__ zbuf,
                      unsigned int* __restrict__ zmaxk, int N) {
    const int lane = threadIdx.x & 31;
    const int wave = (blockIdx.x * blockDim.x + threadIdx.x) >> 5;
    const int n    = lane & 15;          // token column within tile / hidden row within A tile
    const int hi   = lane >> 4;          // upper half-wave flag
    const float msk = hi ? 0.0f : 1.0f;  // zero pad for K=6,7 without divergent loads

    // ---- A = W1^T tiles (16 hidden x 4 feat), loop-invariant ----
    // A layout: lanes 0-15 hold M=lane, K=0(v0),1(v1); lanes 16-31 hold K=2,3.
    const int m0 = n, m1 = n + 16;       // hidden rows for N-tile 0 / 1
    v2f A0t0, A1t0, A0t1, A1t1;
    A0t0.x = W1[(2 * hi + 0) * DHID + m0];
    A0t0.y = W1[(2 * hi + 1) * DHID + m0];
    A1t0.x = W1[4 * DHID + m0] * msk;    // K=4 (lanes<16) / K=6 pad (lanes>=16)
    A1t0.y = W1[5 * DHID + m0] * msk;    // K=5 / K=7 pad
    A0t1.x = W1[(2 * hi + 0) * DHID + m1];
    A0t1.y = W1[(2 * hi + 1) * DHID + m1];
    A1t1.x = W1[4 * DHID + m1] * msk;
    A1t1.y = W1[5 * DHID + m1] * msk;

    // ---- per-lane head params: lane owns hidden {8*hi+v} U {16+8*hi+v}, v=0..7 ----
    float b1a[8], b1b[8], wra[8], wrb[8], wza[8], wzb[8];
#pragma unroll
    for (int v = 0; v < 8; ++v) {
        b1a[v] = b1[8 * hi + v];       b1b[v] = b1[16 + 8 * hi + v];
        wra[v] = wr[8 * hi + v];       wrb[v] = wr[16 + 8 * hi + v];
        wza[v] = wz[8 * hi + v];       wzb[v] = wz[16 + 8 * hi + v];
    }
    const float br = brp[0], bz = bzp[0];

    const int base_tok = wave * TOK_PER_WAVE;

#pragma unroll
    for (int tIdx = 0; tIdx < TILES_PER_WAVE; ++tIdx) {
        const int tok0 = base_tok + tIdx * 16;
        const int tok  = tok0 + n;
        int row = tok; if (row > N - 1) row = N - 1;          // tail clamp
        const float* xr = x + (long)row * FEAT;

        // ---- B = X^T chunk (4 feat x 16 tokens) ----
        // B layout: lanes 0-15 hold N=lane, K=0(v0),1(v1); lanes 16-31 hold K=2,3.
        v2f B0, B1;
        B0.x = xr[2 * hi + 0];
        B0.y = xr[2 * hi + 1];
        B1.x = xr[4] * msk;                                   // K=4 / K=6 pad
        B1.y = xr[5] * msk;                                   // K=5 / K=7 pad

        v8f C0 = {}; v8f C1 = {};
        C0 = __builtin_amdgcn_wmma_f32_16x16x4_f32(false, A0t0, false, B0, (short)0, C0, false, false);
        C0 = __builtin_amdgcn_wmma_f32_16x16x4_f32(false, A1t0, false, B1, (short)0, C0, false, false);
        C1 = __builtin_amdgcn_wmma_f32_16x16x4_f32(false, A0t1, false, B0, (short)0, C1, false, false);
        C1 = __builtin_amdgcn_wmma_f32_16x16x4_f32(false, A1t1, false, B1, (short)0, C1, false, false);

        // D[v] = h[hidden = v+8*hi (+16 for C1)][token = lane&15].
        // Bias + ReLU + both head dot-products, all per-lane.
        float rp = 0.0f, zp = 0.0f;
#pragma unroll
        for (int v = 0; v < 8; ++v) {
            float ha = fmaxf(C0[v] + b1a[v], 0.0f);
            float hb = fmaxf(C1[v] + b1b[v], 0.0f);
            rp += ha * wra[v] + hb * wrb[v];
            zp += ha * wza[v] + hb * wzb[v];
        }
        // lane l covers hidden {8hi..8hi+7, 16+8hi..23+8hi}; lane l^16 covers the rest.
        rp += __shfl_xor(rp, 16, 32);
        zp += __shfl_xor(zp, 16, 32);

        if (hi == 0 && tok < N) {        // 16 writer lanes, consecutive tokens -> coalesced
            float rv = rp + br;
            float zv = zp + bz;
            rbuf[tok] = rv;
            zbuf[tok] = zv;
            atomicMax(&zmaxk[seg[tok]], f32_key(zv));
        }
    }
}

__global__ void k_acc(const float* __restrict__ rbuf, const float* __restrict__ zbuf,
                      const int* __restrict__ seg, const unsigned int* __restrict__ zmaxk,
                      float* __restrict__ num, float* __restrict__ den, int N) {
    int t = blockIdx.x * blockDim.x + threadIdx.x;
    float e = 0.0f, er = 0.0f;
    int s = 0;
    if (t < N) {
        s = seg[t];
        float zm = f32_unkey(zmaxk[s]);
        e  = __expf(zbuf[t] - zm);
        er = e * rbuf[t];
    }
    const int lane = threadIdx.x & 31;
    int s0 = __shfl(s, 0, 32);
    // Segment ids are sorted: waves are almost always segment-uniform.
    bool uni = __all((t < N) && (s == s0));
    if (uni) {
#pragma unroll
        for (int off = 1; off < 32; off <<= 1) {
            e  += __shfl_xor(e,  off, 32);
            er += __shfl_xor(er, off, 32);
        }
        if (lane == 0) {
            atomicAdd(&den[s0], e);
            atomicAdd(&num[s0], er);
        }
    } else if (t < N) {
        atomicAdd(&den[s], e);
        atomicAdd(&num[s], er);
    }
}

__global__ void k_final(const float* __restrict__ num, const float* __restrict__ den,
                        float* __restrict__ out, int G) {
    int g = blockIdx.x * blockDim.x + threadIdx.x;
    if (g < G) {
        float d = den[g];
        out[g] = (d > 0.0f) ? (SCALE_F * num[g] / d) : 0.0f;
    }
}

extern "C" void kernel_launch(void* const* d_in, const int* in_sizes, int n_in,
                              void* d_out, int out_size, void* d_ws, size_t ws_size,
                              hipStream_t stream) {
    const float* x   = (const float*)d_in[0];
    const float* W1  = (const float*)d_in[1];
    const float* b1  = (const float*)d_in[2];
    const float* wr  = (const float*)d_in[3];
    const float* br  = (const float*)d_in[4];
    const float* wz  = (const float*)d_in[5];
    const float* bz  = (const float*)d_in[6];
    const int*   seg = (const int*)d_in[7];
    float* out = (float*)d_out;

    const int N = in_sizes[0] / FEAT;   // 2,097,152
    const int G = out_size;             // 8,192

    // Workspace layout: r[N] | z[N] | zmax_key[G] | num[G] | den[G]
    float*        rbuf  = (float*)d_ws;
    float*        zbuf  = rbuf + N;
    unsigned int* zmaxk = (unsigned int*)(zbuf + N);
    float*        num   = (float*)(zmaxk + G);
    float*        den   = num + G;

    const int BLK = 256;                              // 8 waves / block
    const int tokPerBlk = (BLK / 32) * TOK_PER_WAVE;  // 512 tokens

    k_init<<<(G + BLK - 1) / BLK, BLK, 0, stream>>>(zmaxk, num, den, G);
    k_mlp<<<(N + tokPerBlk - 1) / tokPerBlk, BLK, 0, stream>>>(
        x, W1, b1, wr, br, wz, bz, seg, rbuf, zbuf, zmaxk, N);
    k_acc<<<(N + BLK - 1) / BLK, BLK, 0, stream>>>(rbuf, zbuf, seg, zmaxk, num, den, N);
    k_final<<<(G + BLK - 1) / BLK, BLK, 0, stream>>>(num, den, out, G);
}